// LeadLagAttention_32830730010761
// MI455X (gfx1250) — compile-verified
//
#include <hip/hip_runtime.h>
#include <hip/hip_bf16.h>
#include <math.h>

typedef __attribute__((ext_vector_type(2))) float v2f;
typedef __attribute__((ext_vector_type(8))) float v8f;
typedef int ll_v4i __attribute__((vector_size(16)));   // matches builtin param

#define BATCH 16
#define T_LEN 64
#define N_TOK 2048
#define DH    32
#define SCALE 0.17677669529663687f  /* 1/sqrt(32) */
#define N_TILES (N_TOK / 16)        /* 128 */
#define WAVES 4                     /* n-tiles (waves) per block in kernel 3 */

// ---------------------------------------------------------------------------
// CDNA5 async global->LDS staging (ASYNCcnt path), with portable fallback.
// Device pass with a gfx1250-aware clang emits global_load_async_to_lds_b128
// + s_wait_asynccnt; host pass / older toolchains take the float4 path.
// ---------------------------------------------------------------------------
#if defined(__HIP_DEVICE_COMPILE__) && \
    __has_builtin(__builtin_amdgcn_global_load_async_to_lds_b128) && \
    __has_builtin(__builtin_amdgcn_s_wait_asynccnt)
#define LL_ASYNC 1
#else
#define LL_ASYNC 0
#endif

__device__ __forceinline__ void ll_copy16(const float* __restrict__ g,
                                          float* __restrict__ l) {
#if LL_ASYNC
    __builtin_amdgcn_global_load_async_to_lds_b128(
        (__attribute__((address_space(1))) ll_v4i*)g,
        (__attribute__((address_space(3))) ll_v4i*)l, 0, 0);
#else
    *(float4*)l = *(const float4*)g;
#endif
}

__device__ __forceinline__ void ll_async_wait() {
#if LL_ASYNC
    __builtin_amdgcn_s_wait_asynccnt(0);
#endif
}

// ---------------------------------------------------------------------------
// Kernel 1: per-(b,n) normalize x over T (ddof=1 std + 1e-8), transpose to
// xn[b][n][t] so the WMMA K-dimension (t) is contiguous.
// ---------------------------------------------------------------------------
__global__ __launch_bounds__(256) void ll_normalize_kernel(
    const float* __restrict__ x,   // [B, T, N]
    float* __restrict__ xn)        // [B, N, T]
{
    int tid = blockIdx.x * blockDim.x + threadIdx.x;
    if (tid >= BATCH * N_TOK) return;
    int b = tid >> 11;
    int n = tid & (N_TOK - 1);
    const float* xp = x + (size_t)b * T_LEN * N_TOK + n;

    float vals[T_LEN];
    float s = 0.f;
    #pragma unroll 8
    for (int t = 0; t < T_LEN; ++t) { vals[t] = xp[(size_t)t * N_TOK]; s += vals[t]; }
    float mean = s * (1.0f / T_LEN);
    float ss = 0.f;
    #pragma unroll 8
    for (int t = 0; t < T_LEN; ++t) { float d = vals[t] - mean; ss += d * d; }
    float inv = 1.0f / (sqrtf(ss * (1.0f / (T_LEN - 1))) + 1e-8f);

    float* o = xn + (size_t)tid * T_LEN;
    #pragma unroll 8
    for (int t = 0; t < T_LEN; ++t) o[t] = (vals[t] - mean) * inv;
}

// ---------------------------------------------------------------------------
// Kernel 2: qkv = features @ Wqkv + bqkv. One wave32 per row.
// ---------------------------------------------------------------------------
__global__ __launch_bounds__(256) void ll_qkv_kernel(
    const float* __restrict__ features, // [B*N, 32]
    const float* __restrict__ Wqkv,     // [32, 96]
    const float* __restrict__ bqkv,     // [96]
    float* __restrict__ q, float* __restrict__ k, float* __restrict__ v)
{
    int gtid = blockIdx.x * blockDim.x + threadIdx.x;
    int row  = gtid >> 5;
    int lane = threadIdx.x & 31;
    if (row >= BATCH * N_TOK) return;

    float f = features[(size_t)row * DH + lane];
    float aq = bqkv[lane], ak = bqkv[32 + lane], av = bqkv[64 + lane];
    #pragma unroll
    for (int d = 0; d < DH; ++d) {
        float fd = __shfl(f, d, 32);
        aq = fmaf(fd, Wqkv[d * 96 + lane],      aq);
        ak = fmaf(fd, Wqkv[d * 96 + 32 + lane], ak);
        av = fmaf(fd, Wqkv[d * 96 + 64 + lane], av);
    }
    q[(size_t)row * DH + lane] = aq;
    k[(size_t)row * DH + lane] = ak;
    v[(size_t)row * DH + lane] = av;
}

// ---------------------------------------------------------------------------
// Kernel 3: fused lag-corr + attention, flash style.
// 4 waves / block, each wave owns a 16-row n-tile; the block shares
// double-buffered m-side tiles (xn/k/v) staged with async global->LDS copies.
// Per m-tile: 48 WMMA (3 lag corrs, K over t) + 8 WMMA (QK^T) + 8 WMMA (P@V),
// all v_wmma_f32_16x16x4_f32. The N x N matrices never touch global memory.
// ---------------------------------------------------------------------------
__global__ __launch_bounds__(32 * WAVES) void ll_attn_kernel(
    const float* __restrict__ xn,  // [B, N, T]
    const float* __restrict__ q,   // [B, N, D]
    const float* __restrict__ k,
    const float* __restrict__ v,
    float* __restrict__ att)       // [B, N, D]
{
    // Row strides 68/36/20 floats: conflict-free column access on 64 banks;
    // cols 64..67 of sXn are the zero pad implementing the lag shift and the
    // K = T - lag truncation. 16-byte aligned for b128 LDS traffic.
    __shared__ __align__(16) float sXn[WAVES][16][68]; // per-wave n-side xn
    __shared__ __align__(16) float sQ [WAVES][16][36]; // per-wave q tile
    __shared__ __align__(16) float sP [WAVES][16][20]; // per-wave P transpose
    __shared__ __align__(16) float sXm[2][16][68];     // shared, double-buffered
    __shared__ __align__(16) float sK [2][16][36];
    __shared__ __align__(16) float sV [2][16][36];

    const int tid   = threadIdx.x;       // 0..127
    const int lane  = tid & 31;
    const int w     = tid >> 5;          // wave id 0..3
    const int row16 = lane & 15;         // A-row / B,C,D-col
    const int hi    = lane >> 4;         // lane half
    const int b     = blockIdx.x >> 5;   // 32 blocks per batch
    const int n0    = (((blockIdx.x & 31) * WAVES) + w) * 16;

    const float* xnB = xn + (size_t)b * N_TOK * T_LEN;
    const float* qB  = q  + (size_t)b * N_TOK * DH;
    const float* kB  = k  + (size_t)b * N_TOK * DH;
    const float* vB  = v  + (size_t)b * N_TOK * DH;
    float*       attB = att + (size_t)b * N_TOK * DH;

    // ---- per-wave n-side tiles (plain stores; read only by this wave) ----
    for (int i = lane; i < 256; i += 32) {             // 16 rows x 16 float4
        int r = i >> 4, c = (i & 15) * 4;
        *(float4*)&sXn[w][r][c] = *(const float4*)&xnB[(size_t)(n0 + r) * T_LEN + c];
    }
    if (lane < 16) {
        sXn[w][lane][64] = 0.f; sXn[w][lane][65] = 0.f;
        sXn[w][lane][66] = 0.f; sXn[w][lane][67] = 0.f;
    }
    for (int i = lane; i < 128; i += 32) {             // 16 rows x 8 float4
        int r = i >> 3, c = (i & 7) * 4;
        *(float4*)&sQ[w][r][c] = *(const float4*)&qB[(size_t)(n0 + r) * DH + c];
    }

    // ---- preload loop-invariant Q A-fragments into registers ----
    // (same-wave LDS ops are in order; compiler inserts the dscnt waits)
    v2f aqf[8];
    #pragma unroll
    for (int s = 0; s < 8; ++s) {
        int tb = s * 4 + hi * 2;
        aqf[s].x = sQ[w][row16][tb];
        aqf[s].y = sQ[w][row16][tb + 1];
    }

    // ---- block-cooperative staging of one m-tile (async when available) ----
    auto stage_m = [&](int m0, int buf) {
        #pragma unroll
        for (int i = tid; i < 256; i += 128) {         // xn tile: 256 float4
            int r = i >> 4, c = (i & 15) * 4;
            ll_copy16(&xnB[(size_t)(m0 + r) * T_LEN + c], &sXm[buf][r][c]);
        }
        {
            int r = tid >> 3, c = (tid & 7) * 4;       // k/v tiles: 128 float4
            ll_copy16(&kB[(size_t)(m0 + r) * DH + c], &sK[buf][r][c]);
            ll_copy16(&vB[(size_t)(m0 + r) * DH + c], &sV[buf][r][c]);
        }
    };

    stage_m(0, 0);
    ll_async_wait();
    __syncthreads();

    v8f accA = {}, accB = {};            // attended cols 0-15 / 16-31
    float mrow[8], lrow[8];
    #pragma unroll
    for (int r = 0; r < 8; ++r) { mrow[r] = -INFINITY; lrow[r] = 0.f; }

    for (int mt = 0; mt < N_TILES; ++mt) {
        const int cur = mt & 1;

        if (mt + 1 < N_TILES) stage_m((mt + 1) * 16, cur ^ 1);   // async issue
        if (mt + 2 < N_TILES)                                     // L2 prefetch
            __builtin_prefetch(&xnB[(size_t)(mt + 2) * 16 * T_LEN + tid * 8], 0, 1);

        // ---- score tiles: 3 lag correlations (K over t) + QK^T (K over d)
        v8f c0 = {}, c1 = {}, c3 = {}, cqk = {};
        #pragma unroll
        for (int s = 0; s < 16; ++s) {
            const int tb = s * 4 + hi * 2;   // this lane's K pair base
            v2f bm = { sXm[cur][row16][tb],   sXm[cur][row16][tb + 1] };
            v2f a0 = { sXn[w][row16][tb],     sXn[w][row16][tb + 1] };
            v2f a1 = { sXn[w][row16][tb + 1], sXn[w][row16][tb + 2] };
            v2f a3 = { sXn[w][row16][tb + 3], sXn[w][row16][tb + 4] };
            c0 = __builtin_amdgcn_wmma_f32_16x16x4_f32(false, a0, false, bm, (short)0, c0, false, false);
            c1 = __builtin_amdgcn_wmma_f32_16x16x4_f32(false, a1, false, bm, (short)0, c1, false, false);
            c3 = __builtin_amdgcn_wmma_f32_16x16x4_f32(false, a3, false, bm, (short)0, c3, false, false);
        }
        #pragma unroll
        for (int s = 0; s < 8; ++s) {
            const int tb = s * 4 + hi * 2;
            v2f bk = { sK[cur][row16][tb], sK[cur][row16][tb + 1] };
            cqk = __builtin_amdgcn_wmma_f32_16x16x4_f32(false, aqf[s], false, bk, (short)0, cqk, false, false);
        }

        // ---- combine + online softmax (row r+8*hi lives in VGPR r) ----
        #pragma unroll
        for (int r = 0; r < 8; ++r) {
            float corr = fmaxf(fmaxf(c0[r] * (1.0f / 64.0f), c1[r] * (1.0f / 63.0f)),
                               fmaxf(c3[r] * (1.0f / 61.0f), 0.0f));
            float sv = cqk[r] * SCALE + 0.5f * corr;
            float mx = sv;                          // max over the 16-lane half
            mx = fmaxf(mx, __shfl_xor(mx, 1, 32));
            mx = fmaxf(mx, __shfl_xor(mx, 2, 32));
            mx = fmaxf(mx, __shfl_xor(mx, 4, 32));
            mx = fmaxf(mx, __shfl_xor(mx, 8, 32));
            float mnew  = fmaxf(mrow[r], mx);
            float scale = __expf(mrow[r] - mnew);   // 0 on first tile (-inf)
            float p     = __expf(sv - mnew);
            float ps = p;
            ps += __shfl_xor(ps, 1, 32);
            ps += __shfl_xor(ps, 2, 32);
            ps += __shfl_xor(ps, 4, 32);
            ps += __shfl_xor(ps, 8, 32);
            lrow[r] = lrow[r] * scale + ps;
            mrow[r] = mnew;
            accA[r] *= scale;
            accB[r] *= scale;
            sP[w][r + 8 * hi][row16] = p;           // D-layout -> row-major LDS
        }
        // sP is private to this wave: same-wave LDS ordering suffices, no barrier.

        // ---- acc += P @ V (K = 16 over m) ----
        #pragma unroll
        for (int s = 0; s < 4; ++s) {
            const int tb = s * 4 + hi * 2;
            v2f ap  = { sP[w][row16][tb],        sP[w][row16][tb + 1] };
            v2f bv0 = { sV[cur][tb][row16],      sV[cur][tb + 1][row16] };
            v2f bv1 = { sV[cur][tb][16 + row16], sV[cur][tb + 1][16 + row16] };
            accA = __builtin_amdgcn_wmma_f32_16x16x4_f32(false, ap, false, bv0, (short)0, accA, false, false);
            accB = __builtin_amdgcn_wmma_f32_16x16x4_f32(false, ap, false, bv1, (short)0, accB, false, false);
        }

        ll_async_wait();      // next buffer fully landed in LDS
        __syncthreads();      // all waves done with cur before it is re-staged
    }

    // ---- finalize: attended = acc / l ----
    #pragma unroll
    for (int r = 0; r < 8; ++r) {
        int rn = n0 + r + 8 * hi;
        float invl = 1.0f / lrow[r];
        attB[(size_t)rn * DH + row16]      = accA[r] * invl;
        attB[(size_t)rn * DH + 16 + row16] = accB[r] * invl;
    }
}

// ---------------------------------------------------------------------------
// Kernel 4: out = attended @ Wout + bout; res = out + features; LayerNorm(D).
// ---------------------------------------------------------------------------
__global__ __launch_bounds__(256) void ll_out_ln_kernel(
    const float* __restrict__ att,      // [B*N, 32]
    const float* __restrict__ features, // [B*N, 32]
    const float* __restrict__ Wout,     // [32, 32]
    const float* __restrict__ bout,     // [32]
    const float* __restrict__ gamma,
    const float* __restrict__ beta,
    float* __restrict__ out)            // [B*N, 32]
{
    int gtid = blockIdx.x * blockDim.x + threadIdx.x;
    int row  = gtid >> 5;
    int lane = threadIdx.x & 31;
    if (row >= BATCH * N_TOK) return;

    float a = att[(size_t)row * DH + lane];
    float o = bout[lane];
    #pragma unroll
    for (int d = 0; d < DH; ++d) {
        float ad = __shfl(a, d, 32);
        o = fmaf(ad, Wout[d * DH + lane], o);
    }
    float res = o + features[(size_t)row * DH + lane];

    float s = res;
    s += __shfl_xor(s, 16, 32); s += __shfl_xor(s, 8, 32);
    s += __shfl_xor(s, 4, 32);  s += __shfl_xor(s, 2, 32);
    s += __shfl_xor(s, 1, 32);
    float mu = s * (1.0f / DH);
    float dv = res - mu;
    float sq = dv * dv;
    sq += __shfl_xor(sq, 16, 32); sq += __shfl_xor(sq, 8, 32);
    sq += __shfl_xor(sq, 4, 32);  sq += __shfl_xor(sq, 2, 32);
    sq += __shfl_xor(sq, 1, 32);
    float var = sq * (1.0f / DH);
    float normed = dv * rsqrtf(var + 1e-5f);
    out[(size_t)row * DH + lane] = normed * gamma[lane] + beta[lane];
}

// ---------------------------------------------------------------------------
extern "C" void kernel_launch(void* const* d_in, const int* in_sizes, int n_in,
                              void* d_out, int out_size, void* d_ws, size_t ws_size,
                              hipStream_t stream) {
    const float* x        = (const float*)d_in[0];
    const float* features = (const float*)d_in[1];
    const float* Wqkv     = (const float*)d_in[2];
    const float* bqkv     = (const float*)d_in[3];
    const float* Wout     = (const float*)d_in[4];
    const float* bout     = (const float*)d_in[5];
    const float* gamma    = (const float*)d_in[6];
    const float* beta     = (const float*)d_in[7];

    float* ws  = (float*)d_ws;
    float* xnw = ws;                                   // B*N*T = 2,097,152 f
    float* q   = xnw + (size_t)BATCH * N_TOK * T_LEN;  // B*N*D = 1,048,576 f
    float* k   = q  + (size_t)BATCH * N_TOK * DH;
    float* v   = k  + (size_t)BATCH * N_TOK * DH;
    float* att = v  + (size_t)BATCH * N_TOK * DH;

    const int rows = BATCH * N_TOK;                    // 32768

    ll_normalize_kernel<<<(rows + 255) / 256, 256, 0, stream>>>(x, xnw);
    ll_qkv_kernel<<<(rows * 32) / 256, 256, 0, stream>>>(features, Wqkv, bqkv, q, k, v);
    ll_attn_kernel<<<BATCH * (N_TILES / WAVES), 32 * WAVES, 0, stream>>>(xnw, q, k, v, att);
    ll_out_ln_kernel<<<(rows * 32) / 256, 256, 0, stream>>>(att, features, Wout, bout,
                                                            gamma, beta, (float*)d_out);
}